// NKDGNN_68401649156828
// MI455X (gfx1250) — compile-verified
//
#include <hip/hip_runtime.h>
#include <hip/hip_bf16.h>
#include <math.h>

typedef __attribute__((ext_vector_type(2))) float v2f;
typedef __attribute__((ext_vector_type(8))) float v8f;

#define TPB 256
#define SCAN_T 1024

// ---------- helpers ----------

// float atomic max via signed/unsigned int monotonicity trick (init with -INF).
__device__ __forceinline__ void atomic_max_f32(float* addr, float val) {
  if (val >= 0.0f) atomicMax((int*)addr, __float_as_int(val));
  else             atomicMin((unsigned int*)addr, __float_as_uint(val));
}

// edge_index is [2,E] flat (row0=src, row1=dst); indices >= E are self-loops.
__device__ __forceinline__ int edge_src(const int* __restrict__ ei, int E, int e) {
  return (e < E) ? ei[e] : e - E;
}
__device__ __forceinline__ int edge_dst(const int* __restrict__ ei, int E, int e) {
  return (e < E) ? ei[E + e] : e - E;
}

__global__ void fill_kernel(float* __restrict__ p, float v, int n) {
  int i = blockIdx.x * blockDim.x + threadIdx.x;
  if (i < n) p[i] = v;
}
__global__ void fill_int_kernel(int* __restrict__ p, int v, int n) {
  int i = blockIdx.x * blockDim.x + threadIdx.x;
  if (i < n) p[i] = v;
}
__global__ void copy_int_kernel(const int* __restrict__ a, int* __restrict__ b, int n) {
  int i = blockIdx.x * blockDim.x + threadIdx.x;
  if (i < n) b[i] = a[i];
}

// ---------- CSR construction (incoming edges per dst node) ----------
__global__ void deg_count_kernel(const int* __restrict__ ei, int* __restrict__ deg,
                                 int E, int n) {
  int e = blockIdx.x * blockDim.x + threadIdx.x;
  if (e >= E + n) return;
  atomicAdd(&deg[edge_dst(ei, E, e)], 1);
}

// single-workgroup exclusive scan (n ~ 50k, runs once)
__global__ void scan_kernel(const int* __restrict__ deg, int* __restrict__ offs, int n) {
  __shared__ int sh[SCAN_T];
  __shared__ int carry;
  if (threadIdx.x == 0) carry = 0;
  __syncthreads();
  for (int base = 0; base < n; base += SCAN_T) {
    int i = base + (int)threadIdx.x;
    int v = (i < n) ? deg[i] : 0;
    sh[threadIdx.x] = v;
    __syncthreads();
    for (int off = 1; off < SCAN_T; off <<= 1) {
      int t = (threadIdx.x >= (unsigned)off) ? sh[threadIdx.x - off] : 0;
      __syncthreads();
      sh[threadIdx.x] += t;
      __syncthreads();
    }
    if (i < n) offs[i] = carry + sh[threadIdx.x] - v;   // exclusive
    __syncthreads();
    if (threadIdx.x == 0) carry += sh[SCAN_T - 1];
    __syncthreads();
  }
  if (threadIdx.x == 0) offs[n] = carry;
}

__global__ void scatter_kernel(const int* __restrict__ ei, int* __restrict__ cursor,
                               int* __restrict__ eid, int E, int n) {
  int e = blockIdx.x * blockDim.x + threadIdx.x;
  if (e >= E + n) return;
  int pos = atomicAdd(&cursor[edge_dst(ei, E, e)], 1);
  eid[pos] = e;
}

// ---------- dense GEMM: OUT[M,Ncols] = X[M,K] @ W[K,Ncols] via V_WMMA_F32_16X16X4_F32
// One wave per 16x16 tile; dual accumulators break the WMMA->WMMA RAW chain.
__global__ void gemm_wmma_f32(const float* __restrict__ X, const float* __restrict__ W,
                              float* __restrict__ OUT, int M, int K, int Ncols) {
  const int wave = (blockIdx.x * blockDim.x + threadIdx.x) >> 5;
  const int lane = threadIdx.x & 31;
  const int tilesM = (M + 15) >> 4;
  const int tilesN = Ncols >> 4;
  if (wave >= tilesM * tilesN) return;               // uniform per wave
  const int tm = wave / tilesN, tn = wave % tilesN;
  const int row0 = tm << 4, col0 = tn << 4;
  const int half = lane >> 4;                         // 0 -> K{0,1}, 1 -> K{2,3}
  const int l16  = lane & 15;
  int arow = row0 + l16; if (arow >= M) arow = M - 1; // data clamp, EXEC stays full
  const float* __restrict__ xrow = X + (size_t)arow * K;
  const float* __restrict__ wcol = W + col0 + l16;

  v8f acc0 = {}, acc1 = {};
  for (int k = 0; k < K; k += 8) {
    {
      const int ka = k + (half << 1);
      v2f a = *(const v2f*)(xrow + ka);               // 8B-aligned (ka even)
      v2f b;
      b[0] = wcol[(size_t)ka * Ncols];
      b[1] = wcol[(size_t)(ka + 1) * Ncols];
      acc0 = __builtin_amdgcn_wmma_f32_16x16x4_f32(false, a, false, b, (short)0, acc0,
                                                   false, false);
    }
    {
      const int ka = k + 4 + (half << 1);
      v2f a = *(const v2f*)(xrow + ka);
      v2f b;
      b[0] = wcol[(size_t)ka * Ncols];
      b[1] = wcol[(size_t)(ka + 1) * Ncols];
      acc1 = __builtin_amdgcn_wmma_f32_16x16x4_f32(false, a, false, b, (short)0, acc1,
                                                   false, false);
    }
  }
  v8f acc = acc0 + acc1;
  #pragma unroll
  for (int r = 0; r < 8; ++r) {                        // D: VGPR r -> M = r + 8*half
    const int m = row0 + r + (half << 3);
    if (m < M) OUT[(size_t)m * Ncols + col0 + l16] = acc[r];
  }
}

// ---------- per-node attention coefficients: s = <h, a_src>, d = <h, a_dst>
__global__ void attn_sd_kernel(const float* __restrict__ h, const float* __restrict__ asrc,
                               const float* __restrict__ adst, float* __restrict__ s,
                               float* __restrict__ d, int n, int heads, int ch) {
  int idx = blockIdx.x * blockDim.x + threadIdx.x;
  if (idx >= n * heads) return;
  int node = idx / heads, hh = idx % heads;
  const float* hp = h + (size_t)node * heads * ch + hh * ch;
  const float* as = asrc + hh * ch;
  const float* ad = adst + hh * ch;
  float ss = 0.f, dd = 0.f;
  for (int c = 0; c < ch; ++c) { float v = hp[c]; ss += v * as[c]; dd += v * ad[c]; }
  s[idx] = ss; d[idx] = dd;
}

// ---------- segment softmax stats via CSR gather (no atomics)
__global__ void node_softmax_kernel(const int* __restrict__ ei, const int* __restrict__ offs,
                                    const int* __restrict__ eid, const float* __restrict__ s,
                                    const float* __restrict__ d, float* __restrict__ m,
                                    float* __restrict__ den, int E, int n, int heads) {
  int idx = blockIdx.x * blockDim.x + threadIdx.x;
  if (idx >= n * heads) return;
  int node = idx / heads, hh = idx % heads;
  int beg = offs[node], end = offs[node + 1];
  float dd = d[idx];
  float mx = -INFINITY;
  for (int j = beg; j < end; ++j) {
    int sn = edge_src(ei, E, eid[j]);
    float ev = s[sn * heads + hh] + dd;
    ev = ev > 0.f ? ev : 0.2f * ev;
    mx = fmaxf(mx, ev);
  }
  float sum = 0.f;
  for (int j = beg; j < end; ++j) {
    int sn = edge_src(ei, E, eid[j]);
    float ev = s[sn * heads + hh] + dd;
    ev = ev > 0.f ? ev : 0.2f * ev;
    sum += __expf(ev - mx);
  }
  m[idx] = mx; den[idx] = sum;
}

// ---------- aggregation via CSR gather: out[node] = relu(bias + sum alpha*h[src])
// one thread per (node, 4-channel chunk); registers accumulate, single float4 store.
__global__ void gather_aggr_kernel(const int* __restrict__ ei, const int* __restrict__ offs,
                                   const int* __restrict__ eid, const float* __restrict__ s,
                                   const float* __restrict__ d, const float* __restrict__ m,
                                   const float* __restrict__ den, const float* __restrict__ h,
                                   const float* __restrict__ bias, float* __restrict__ out,
                                   int E, int n, int heads, int ch) {
  int t = blockIdx.x * blockDim.x + threadIdx.x;
  int chunks = (heads * ch) >> 2;
  if (t >= n * chunks) return;
  int node = t / chunks, q = t % chunks;
  int hh = (q << 2) / ch;
  int beg = offs[node], end = offs[node + 1];
  float dd = d[node * heads + hh];
  float mx = m[node * heads + hh];
  float invden = 1.0f / den[node * heads + hh];
  float ax = 0.f, ay = 0.f, az = 0.f, aw = 0.f;
  for (int j = beg; j < end; ++j) {
    int sn = edge_src(ei, E, eid[j]);
    float ev = s[sn * heads + hh] + dd;
    ev = ev > 0.f ? ev : 0.2f * ev;
    float alpha = __expf(ev - mx) * invden;
    const float4 hv = *(const float4*)(h + (size_t)sn * heads * ch + (q << 2));
    ax += alpha * hv.x; ay += alpha * hv.y; az += alpha * hv.z; aw += alpha * hv.w;
  }
  int cbase = q << 2;
  float4 r;
  r.x = ax + bias[cbase + 0]; r.x = r.x > 0.f ? r.x : 0.f;
  r.y = ay + bias[cbase + 1]; r.y = r.y > 0.f ? r.y : 0.f;
  r.z = az + bias[cbase + 2]; r.z = r.z > 0.f ? r.z : 0.f;
  r.w = aw + bias[cbase + 3]; r.w = r.w > 0.f ? r.w : 0.f;
  *(float4*)(out + (size_t)node * heads * ch + cbase) = r;
}

// ---------- pooling attention score + entity sigmoid
__global__ void node_score_kernel(const float* __restrict__ h2, const float* __restrict__ aw,
                                  const float* __restrict__ ab, const float* __restrict__ fw,
                                  const float* __restrict__ fb, float* __restrict__ score,
                                  float* __restrict__ ent, int n, int ch) {
  int i = blockIdx.x * blockDim.x + threadIdx.x;
  if (i >= n) return;
  const float* hp = h2 + (size_t)i * ch;
  float sa = 0.f, sf = 0.f;
  for (int c = 0; c < ch; ++c) { float v = hp[c]; sa += v * aw[c]; sf += v * fw[c]; }
  score[i] = sa + ab[0];
  float z = sf + fb[0];
  ent[i] = 1.0f / (1.0f + __expf(-z));
}

__global__ void reduce_max_kernel(const float* __restrict__ s, float* __restrict__ gmax, int n) {
  __shared__ float red[TPB];
  int tid = threadIdx.x;
  float v = -INFINITY;
  for (int i = blockIdx.x * blockDim.x + tid; i < n; i += gridDim.x * blockDim.x)
    v = fmaxf(v, s[i]);
  red[tid] = v;
  __syncthreads();
  for (int off = TPB / 2; off > 0; off >>= 1) {
    if (tid < off) red[tid] = fmaxf(red[tid], red[tid + off]);
    __syncthreads();
  }
  if (tid == 0) atomic_max_f32(gmax, red[0]);
}

__global__ void reduce_sumexp_kernel(const float* __restrict__ s, const float* __restrict__ gmax,
                                     float* __restrict__ gsum, int n) {
  __shared__ float red[TPB];
  int tid = threadIdx.x;
  float mx = *gmax, v = 0.f;
  for (int i = blockIdx.x * blockDim.x + tid; i < n; i += gridDim.x * blockDim.x)
    v += __expf(s[i] - mx);
  red[tid] = v;
  __syncthreads();
  for (int off = TPB / 2; off > 0; off >>= 1) {
    if (tid < off) red[tid] += red[tid + off];
    __syncthreads();
  }
  if (tid == 0) atomicAdd(gsum, red[0]);
}

__global__ void global_rep_kernel(const float* __restrict__ h2, const float* __restrict__ score,
                                  const float* __restrict__ gred, float* __restrict__ rep,
                                  int n, int ch) {
  __shared__ float acc[64];
  int tid = threadIdx.x;
  if (tid < ch) acc[tid] = 0.f;
  __syncthreads();
  float mx = gred[0], sm = gred[1];
  for (int i = blockIdx.x * blockDim.x + tid; i < n; i += gridDim.x * blockDim.x) {
    float w = __expf(score[i] - mx) / sm;
    const float* hp = h2 + (size_t)i * ch;
    for (int c = 0; c < ch; ++c) atomicAdd(&acc[c], w * hp[c]);
  }
  __syncthreads();
  if (tid < ch) atomicAdd(&rep[tid], acc[tid]);
}

// ---------- launcher ----------
extern "C" void kernel_launch(void* const* d_in, const int* in_sizes, int n_in,
                              void* d_out, int out_size, void* d_ws, size_t ws_size,
                              hipStream_t stream) {
  (void)n_in; (void)out_size; (void)ws_size;
  const float* x   = (const float*)d_in[0];
  const int*   ei  = (const int*)d_in[1];
  const float* W1  = (const float*)d_in[2];
  const float* as1 = (const float*)d_in[3];
  const float* ad1 = (const float*)d_in[4];
  const float* b1  = (const float*)d_in[5];
  const float* W2  = (const float*)d_in[6];
  const float* as2 = (const float*)d_in[7];
  const float* ad2 = (const float*)d_in[8];
  const float* b2  = (const float*)d_in[9];
  const float* aw  = (const float*)d_in[10];
  const float* ab  = (const float*)d_in[11];
  const float* fw  = (const float*)d_in[12];
  const float* fb  = (const float*)d_in[13];
  float* out = (float*)d_out;

  const int IN = 256, H = 4, C = 64, HC = H * C;
  const int N = in_sizes[0] / IN;
  const int E = in_sizes[1] / 2;
  const int Etot = E + N;

  float* ws = (float*)d_ws;
  size_t p = 0;
  float* h1pre = ws + p; p += (size_t)N * HC;
  float* h1    = ws + p; p += (size_t)N * HC;
  float* s1    = ws + p; p += (size_t)N * H;
  float* d1    = ws + p; p += (size_t)N * H;
  float* m1    = ws + p; p += (size_t)N * H;
  float* den1  = ws + p; p += (size_t)N * H;
  float* h2pre = ws + p; p += (size_t)N * C;
  float* h2    = ws + p; p += (size_t)N * C;
  float* s2    = ws + p; p += (size_t)N;
  float* d2    = ws + p; p += (size_t)N;
  float* m2    = ws + p; p += (size_t)N;
  float* den2  = ws + p; p += (size_t)N;
  float* score = ws + p; p += (size_t)N;
  float* gred  = ws + p; p += 2;
  int* deg    = (int*)(ws + p); p += (size_t)N;
  int* offs   = (int*)(ws + p); p += (size_t)N + 1;
  int* cursor = (int*)(ws + p); p += (size_t)N;
  int* eid    = (int*)(ws + p); p += (size_t)Etot;

  auto blocks = [](long long t) { return (int)((t + TPB - 1) / TPB); };
  const int wpb = TPB / 32;

  // ---- build incoming-edge CSR (shared by both layers) ----
  fill_int_kernel<<<blocks((long long)N), TPB, 0, stream>>>(deg, 0, N);
  deg_count_kernel<<<blocks((long long)Etot), TPB, 0, stream>>>(ei, deg, E, N);
  scan_kernel<<<1, SCAN_T, 0, stream>>>(deg, offs, N);
  copy_int_kernel<<<blocks((long long)N), TPB, 0, stream>>>(offs, cursor, N);
  scatter_kernel<<<blocks((long long)Etot), TPB, 0, stream>>>(ei, cursor, eid, E, N);

  // ---- GAT layer 1 (heads=4, ch=64, concat) ----
  {
    int tiles = ((N + 15) / 16) * (HC / 16);
    gemm_wmma_f32<<<(tiles + wpb - 1) / wpb, TPB, 0, stream>>>(x, W1, h1pre, N, IN, HC);
  }
  attn_sd_kernel<<<blocks((long long)N * H), TPB, 0, stream>>>(h1pre, as1, ad1, s1, d1, N, H, C);
  node_softmax_kernel<<<blocks((long long)N * H), TPB, 0, stream>>>(ei, offs, eid, s1, d1,
                                                                    m1, den1, E, N, H);
  gather_aggr_kernel<<<blocks((long long)N * (HC / 4)), TPB, 0, stream>>>(
      ei, offs, eid, s1, d1, m1, den1, h1pre, b1, h1, E, N, H, C);

  // ---- GAT layer 2 (heads=1, ch=64, mean==identity) ----
  {
    int tiles = ((N + 15) / 16) * (C / 16);
    gemm_wmma_f32<<<(tiles + wpb - 1) / wpb, TPB, 0, stream>>>(h1, W2, h2pre, N, HC, C);
  }
  attn_sd_kernel<<<blocks((long long)N), TPB, 0, stream>>>(h2pre, as2, ad2, s2, d2, N, 1, C);
  node_softmax_kernel<<<blocks((long long)N), TPB, 0, stream>>>(ei, offs, eid, s2, d2,
                                                                m2, den2, E, N, 1);
  gather_aggr_kernel<<<blocks((long long)N * (C / 4)), TPB, 0, stream>>>(
      ei, offs, eid, s2, d2, m2, den2, h2pre, b2, h2, E, N, 1, C);

  // ---- pooling softmax over nodes + entity scores ----
  fill_kernel<<<1, TPB, 0, stream>>>(gred, -INFINITY, 1);
  fill_kernel<<<1, TPB, 0, stream>>>(gred + 1, 0.f, 1);
  fill_kernel<<<1, TPB, 0, stream>>>(out + N, 0.f, C);
  node_score_kernel<<<blocks((long long)N), TPB, 0, stream>>>(h2, aw, ab, fw, fb, score, out, N, C);
  reduce_max_kernel<<<64, TPB, 0, stream>>>(score, gred, N);
  reduce_sumexp_kernel<<<64, TPB, 0, stream>>>(score, gred, gred + 1, N);
  global_rep_kernel<<<64, TPB, 0, stream>>>(h2, score, gred, out + N, N, C);
}